// SignalGenerator_41558103556571
// MI455X (gfx1250) — compile-verified
//
#include <hip/hip_runtime.h>
#include <math.h>

typedef __attribute__((ext_vector_type(2))) float v2f;
typedef __attribute__((ext_vector_type(8))) float v8f;

#define NROW 32
#define NBLK 2048                 // input samples per row == output 256-blocks per row
#define TLEN (NBLK * 256)         // 524288 output samples per row
#define PHASE_C 3.9269908169872414e-4f   // 2*pi/16000

// ---------------------------------------------------------------------------
// Kernel 1: per-row hierarchical scan of per-block phase increments.
// One wave (32 lanes) per row. The 16-wide inclusive scan is done as
// S = L * X with V_WMMA_F32_16X16X4_F32 (L = lower-triangular ones),
// 4 chained K=4 WMMAs per 16x16 tile, 8 tiles = 2048 values per row.
// Cross-column (128 columns) scan via __shfl_up wave scan.
// Output: phase0[row][k] = exclusive prefix of block sums (phase at block start).
// ---------------------------------------------------------------------------
__global__ __launch_bounds__(32)
void sg_scan_kernel(const float* __restrict__ pitch, float* __restrict__ phase0) {
    const int row  = blockIdx.x;
    const int lane = threadIdx.x;          // 0..31, full wave (EXEC all ones at WMMA)

    __shared__ float p[NBLK];
    __shared__ float bs[NBLK];
    __shared__ float coltot[128];
    __shared__ float offs[128];

    // Load the pitch row (8 KB) into LDS, 128-bit loads.
    {
        const float4* src = (const float4*)(pitch + (size_t)row * NBLK);
        float4* dst = (float4*)p;
        for (int i = lane; i < NBLK / 4; i += 32) dst[i] = src[i];
    }
    __syncthreads();

    // Per-256-sample-block phase increment (closed form of the piecewise-linear
    // upsample cumsum): c * (32*p[k-1] + 192*p[k] + 32*p[k+1]), edge-clamped.
    for (int i = 0; i < NBLK / 32; ++i) {
        int k = lane + 32 * i;
        float pa = p[k == 0 ? 0 : k - 1];
        float pb = p[k];
        float pc = p[k == NBLK - 1 ? NBLK - 1 : k + 1];
        bs[k] = PHASE_C * (32.0f * pa + 192.0f * pb + 32.0f * pc);
    }
    __syncthreads();

    const int n  = lane & 15;   // column (N) for B/C/D; also row (M) for A fragments
    const int hi = lane >> 4;   // half-wave selector per ISA f32 layouts

    // 8 tiles of 256 values; tile t, column c holds bs[t*256 + 16*c .. +15].
    v8f S[8];
    #pragma unroll
    for (int t = 0; t < 8; ++t) {
        v8f acc = {};
        #pragma unroll
        for (int tt = 0; tt < 4; ++tt) {
            // A = L[:, 4tt .. 4tt+3]  (16x4, f32): lane holds M=n, K = 4tt+2*hi (+1)
            int kcol = 4 * tt + 2 * hi;
            v2f a, b;
            a.x = (n >= kcol)     ? 1.0f : 0.0f;
            a.y = (n >= kcol + 1) ? 1.0f : 0.0f;
            // B = X[4tt .. 4tt+3, :] (4x16, f32): lane holds N=n, same K split
            b.x = bs[t * 256 + 16 * n + kcol];
            b.y = bs[t * 256 + 16 * n + kcol + 1];
            acc = __builtin_amdgcn_wmma_f32_16x16x4_f32(
                false, a, false, b, (short)0, acc, false, false);
        }
        S[t] = acc;
        // Column totals live in row M=15 = VGPR 7 of the hi==1 half.
        if (hi == 1) coltot[t * 16 + n] = acc[7];
    }
    __syncthreads();

    // Exclusive scan of the 128 column totals: 4 per lane + wave scan.
    float c0 = coltot[4 * lane + 0];
    float s0 = c0;
    float s1 = s0 + coltot[4 * lane + 1];
    float s2 = s1 + coltot[4 * lane + 2];
    float s3 = s2 + coltot[4 * lane + 3];
    float acc = s3;
    #pragma unroll
    for (int d = 1; d < 32; d <<= 1) {
        float up = __shfl_up(acc, d, 32);
        if (lane >= d) acc += up;
    }
    float excl = acc - s3;
    offs[4 * lane + 0] = excl;
    offs[4 * lane + 1] = excl + s0;
    offs[4 * lane + 2] = excl + s1;
    offs[4 * lane + 3] = excl + s2;
    __syncthreads();

    // Emit exclusive block prefix: P_excl[k] = (S[i,c] + offs[col]) - bs[k].
    float* out = phase0 + (size_t)row * NBLK;
    #pragma unroll
    for (int t = 0; t < 8; ++t) {
        float off = offs[t * 16 + n];
        #pragma unroll
        for (int r = 0; r < 8; ++r) {
            int i = r + 8 * hi;                 // C/D: M = vgpr + 8*hi
            int k = t * 256 + 16 * n + i;
            out[k] = S[t][r] + off - bs[k];
        }
    }
}

// ---------------------------------------------------------------------------
// Kernel 2: streaming synthesis. Each thread produces 4 consecutive output
// samples (float4 noise in, float4 out). Within-block cumsum is closed-form
// (quadratic in r), so no serial dependency remains. 16384 blocks x 256 thr.
// ---------------------------------------------------------------------------
__global__ __launch_bounds__(256)
void sg_synth_kernel(const float* __restrict__ pitch,
                     const float* __restrict__ pamp,
                     const float* __restrict__ apamp,
                     const float* __restrict__ noise,
                     const float* __restrict__ phase0,
                     float* __restrict__ out) {
    const int sub = threadIdx.x >> 6;          // which of 4 output-blocks
    const int t64 = threadIdx.x & 63;          // lane within output-block group
    const int g   = blockIdx.x * 4 + sub;      // global 256-block id
    const int b   = g >> 11;                   // row
    const int k   = g & (NBLK - 1);            // block within row
    const int r0  = t64 * 4;                   // first sample of this thread

    const int km = (k == 0) ? 0 : k - 1;
    const int kp = (k == NBLK - 1) ? NBLK - 1 : k + 1;
    const size_t rb = (size_t)b * NBLK;

    const float pa = pitch[rb + km], pb = pitch[rb + k], pc = pitch[rb + kp];
    const float aa = pamp [rb + km], ab = pamp [rb + k], ac = pamp [rb + kp];
    const float qa = apamp[rb + km], qb = apamp[rb + k], qc = apamp[rb + kp];
    const float ph0 = phase0[rb + k];

    const size_t base = (size_t)b * TLEN + (size_t)k * 256 + r0;
    const float4 nz = *(const float4*)(noise + base);
    float no[4] = {nz.x, nz.y, nz.z, nz.w};
    float o[4];

    #pragma unroll
    for (int j = 0; j < 4; ++j) {
        const int r = r0 + j;
        float ph, amp, ap;
        if (r < 128) {                         // interp between p[k-1], p[k]
            const float rp = (float)(r + 1);
            const float W  = rp * (0.5f + rp * (1.0f / 512.0f));
            ph  = ph0 + PHASE_C * (rp * pa + W * (pb - pa));
            const float w = 0.5f + ((float)r + 0.5f) * (1.0f / 256.0f);
            amp = aa + w * (ab - aa);
            ap  = qa + w * (qb - qa);
        } else {                               // interp between p[k], p[k+1]
            const float q  = (float)(r - 128);
            const float qp = q + 1.0f;
            const float V  = qp * qp * (1.0f / 512.0f);
            ph  = ph0 + PHASE_C * ((32.0f * pa + 96.0f * pb) + qp * pb + V * (pc - pb));
            const float w = (q + 0.5f) * (1.0f / 256.0f);
            amp = ab + w * (ac - ab);
            ap  = qb + w * (qc - qb);
        }
        o[j] = amp * __sinf(ph) + ap * no[j];
    }

    float4 ov = {o[0], o[1], o[2], o[3]};
    *(float4*)(out + base) = ov;
}

extern "C" void kernel_launch(void* const* d_in, const int* in_sizes, int n_in,
                              void* d_out, int out_size, void* d_ws, size_t ws_size,
                              hipStream_t stream) {
    const float* pitch = (const float*)d_in[0];   // [32, 2048]
    const float* pamp  = (const float*)d_in[1];   // [32, 2048]
    const float* apamp = (const float*)d_in[2];   // [32, 2048]
    const float* noise = (const float*)d_in[3];   // [32, 524288]
    float* out    = (float*)d_out;                // [32, 524288]
    float* phase0 = (float*)d_ws;                 // 32*2048 floats = 256 KB scratch

    sg_scan_kernel<<<NROW, 32, 0, stream>>>(pitch, phase0);
    sg_synth_kernel<<<(NROW * NBLK) / 4, 256, 0, stream>>>(
        pitch, pamp, apamp, noise, phase0, out);
}